// DecoderLayer_7765300871321
// MI455X (gfx1250) — compile-verified
//
#include <hip/hip_runtime.h>

// Problem constants (from reference)
#define IN_F   512
#define OUT_F  1024
#define KW     (IN_F + OUT_F - 1)   // 1535 = weight row length
#define BATCH  8192
#define ROWS   16                   // batch rows per workgroup (WMMA M)
#define NWAVES 8                    // 256 threads = 8 wave32
#define BLK    16                   // scan step block (WMMA K for rank-16 update)
#define NTILES (OUT_F / 16)         // 64 column tiles
#define ACC_STRIDE (OUT_F + 4)      // 1028: pad so row stride % 64 banks == 4
#define DSS_STRIDE 18               // sample tile [row][t] pad: 18*lane % 64 all distinct

typedef __attribute__((ext_vector_type(2))) float v2f;
typedef __attribute__((ext_vector_type(4))) float v4f;
typedef __attribute__((ext_vector_type(8))) float v8f;

// Packed transposed layout: element (k, n) lives at [(k>>1)*2*OUT_F + n*2 + (k&1)].
// A (k,k+1) pair for one column n is one aligned 8-byte unit; 16 consecutive
// lanes (consecutive n) read 128 contiguous bytes -> one coalesced b64/lane.
__device__ __forceinline__ size_t pk(int k, int n) {
  return (size_t)(k >> 1) * (2 * OUT_F) + (size_t)n * 2 + (k & 1);
}

// ---------------------------------------------------------------------------
// Prep: pack Wx^T (512x1024) and Wo^T (1024x1024, row 1023 zeroed) into d_ws.
// ---------------------------------------------------------------------------
__global__ void pack_weights_kernel(const float* __restrict__ weight,
                                    float* __restrict__ woT,   // 1024*1024
                                    float* __restrict__ wxT) { // 512*1024
  int idx = blockIdx.x * blockDim.x + threadIdx.x;  // 0 .. 1536*1024-1
  int j = idx >> 10;
  int i = idx & (OUT_F - 1);
  if (j < IN_F) {
    wxT[pk(j, i)] = weight[i * KW + j];
  } else {
    int jj = j - IN_F;                              // feedback step 0..1023
    float v = (jj < OUT_F - 1) ? weight[i * KW + IN_F + jj] : 0.0f;
    woT[pk(jj, i)] = v;
  }
}

// ---------------------------------------------------------------------------
// Main NADE kernel: one workgroup per 16 batch rows, all 1024 steps.
// ---------------------------------------------------------------------------
__launch_bounds__(256)
__global__ void nade_scan_kernel(const float* __restrict__ x,
                                 const float* __restrict__ bias,
                                 const float* __restrict__ u,
                                 const float* __restrict__ woT,
                                 const float* __restrict__ wxT,
                                 float* __restrict__ out_samples,
                                 float* __restrict__ out_logits) {
  __shared__ float acc[ROWS * ACC_STRIDE];   // running logits, all 1024 cols
  __shared__ float dsS[ROWS * DSS_STRIDE];   // [row][t] samples of current block
  __shared__ float dsW[BLK * BLK];           // [t][t2] diagonal Wo block

  const int tid  = threadIdx.x;
  const int lane = tid & 31;
  const int wid  = tid >> 5;
  const int lo   = lane & 15;    // N (or M) index within fragment
  const int hi   = lane >> 4;    // selects K pair {0,1} vs {2,3} per ISA layout
  const int rowBase = blockIdx.x * ROWS;

  // ---- Phase A: acc = x_tile @ Wx^T + bias  (f32 WMMA 16x16x4) -----------
  {
    const int n0base = wid * 128;            // each wave owns 128 output cols
    v8f c[8] = {};
    const float* xrow = x + (size_t)(rowBase + lo) * IN_F;
    for (int k0 = 0; k0 < IN_F; k0 += 4) {
      const int kk = k0 + hi * 2;
      v2f a = *(const v2f*)(xrow + kk);                 // A[M=lo][K=kk,kk+1]
      const float* bb = wxT + (size_t)(kk >> 1) * (2 * OUT_F);
      #pragma unroll
      for (int nt = 0; nt < 8; ++nt) {
        const int n = n0base + nt * 16 + lo;
        v2f b = *(const v2f*)(bb + n * 2);              // B[K=kk,kk+1][N=n]
        c[nt] = __builtin_amdgcn_wmma_f32_16x16x4_f32(
            false, a, false, b, (short)0, c[nt], false, false);
      }
    }
    #pragma unroll
    for (int nt = 0; nt < 8; ++nt) {
      const int n = n0base + nt * 16 + lo;
      const float bv = bias[n];
      #pragma unroll
      for (int r = 0; r < 8; ++r)
        acc[(r + hi * 8) * ACC_STRIDE + n] = c[nt][r] + bv;
    }
  }
  __syncthreads();

  // ---- Scan over 64 blocks of 16 steps -----------------------------------
  for (int g = 0; g < NTILES; ++g) {
    const int c0 = g * BLK;

    if (wid == 0) {
      // Stage diagonal 16x16 Wo block (uniform values) into LDS.
      #pragma unroll
      for (int e = 0; e < 8; ++e) {
        const int idx = lane * 8 + e;
        dsW[idx] = woT[pk(c0 + (idx >> 4), c0 + (idx & 15))];
      }
      const int r = lo;                     // lanes 16-31 shadow rows 0-15
      v4f lc[4], uvv[4], sv[4], lg[4];
      const float* urow = u + (size_t)(rowBase + r) * OUT_F + c0;
      #pragma unroll
      for (int q = 0; q < 4; ++q) {
        lc[q]  = *(const v4f*)(&acc[r * ACC_STRIDE + c0 + q * 4]);
        uvv[q] = *(const v4f*)(urow + q * 4);
      }
      // 16 sequential sampling steps, in-register triangular updates.
      #pragma unroll
      for (int t = 0; t < BLK; ++t) {
        const float logit = lc[t >> 2][t & 3];
        const float sg  = 1.0f / (1.0f + __expf(-logit));   // v_exp_f32
        const float smp = (uvv[t >> 2][t & 3] < sg) ? 1.0f : 0.0f;
        sv[t >> 2][t & 3] = smp;
        lg[t >> 2][t & 3] = logit;
        dsS[r * DSS_STRIDE + t] = smp;   // dup write by lanes 16-31: same value
        #pragma unroll
        for (int t2 = t + 1; t2 < BLK; ++t2)
          lc[t2 >> 2][t2 & 3] += smp * dsW[t * 16 + t2];
      }
      if (lane < 16) {                    // flush samples/logits as b128
        float* so = out_samples + (size_t)(rowBase + r) * OUT_F + c0;
        float* lo2 = out_logits + (size_t)(rowBase + r) * OUT_F + c0;
        #pragma unroll
        for (int q = 0; q < 4; ++q) {
          *(v4f*)(so + q * 4)  = sv[q];
          *(v4f*)(lo2 + q * 4) = lg[q];
        }
      }
    }
    __syncthreads();

    // ---- Phase C: rank-16 WMMA update of all future columns --------------
    // acc[:, future] += S(16x16) @ WoT[c0:c0+16, future]
    for (int nt = g + 1 + wid; nt < NTILES; nt += NWAVES) {
      const int n = nt * 16 + lo;
      v8f c;
      #pragma unroll
      for (int r = 0; r < 8; ++r) c[r] = acc[(r + hi * 8) * ACC_STRIDE + n];
      #pragma unroll
      for (int kb = 0; kb < BLK; kb += 4) {
        const int kk = kb + hi * 2;
        v2f a = *(const v2f*)(&dsS[lo * DSS_STRIDE + kk]);  // S[row][kk,kk+1]
        v2f b = *(const v2f*)(woT + pk(c0 + kk, n));        // WoT[c0+kk..+1][n]
        c = __builtin_amdgcn_wmma_f32_16x16x4_f32(
            false, a, false, b, (short)0, c, false, false);
      }
      #pragma unroll
      for (int r = 0; r < 8; ++r) acc[(r + hi * 8) * ACC_STRIDE + n] = c[r];
      if (nt + NWAVES < NTILES)  // speculative prefetch of next tile's weights
        __builtin_prefetch(woT + pk(c0, (nt + NWAVES) * 16), 0, 0);
    }
    __syncthreads();
  }
}

// ---------------------------------------------------------------------------
extern "C" void kernel_launch(void* const* d_in, const int* in_sizes, int n_in,
                              void* d_out, int out_size, void* d_ws, size_t ws_size,
                              hipStream_t stream) {
  (void)in_sizes; (void)n_in; (void)out_size; (void)ws_size;
  const float* x      = (const float*)d_in[0];
  const float* weight = (const float*)d_in[1];
  const float* bias   = (const float*)d_in[2];
  const float* u      = (const float*)d_in[3];

  float* out_samples = (float*)d_out;                         // (B, OUT_F)
  float* out_logits  = out_samples + (size_t)BATCH * OUT_F;   // (B, OUT_F)
  float* woT         = (float*)d_ws;                          // 4 MB
  float* wxT         = woT + (size_t)OUT_F * OUT_F;           // +2 MB

  pack_weights_kernel<<<((IN_F + OUT_F) * OUT_F) / 256, 256, 0, stream>>>(
      weight, woT, wxT);
  nade_scan_kernel<<<BATCH / ROWS, 256, 0, stream>>>(
      x, bias, u, woT, wxT, out_samples, out_logits);
}